// JaxFusedMoE_59279138619866
// MI455X (gfx1250) — compile-verified
//
#include <hip/hip_runtime.h>

// ---------------- problem constants (from reference) ----------------
#define T_TOK   1024
#define HDIM    1024
#define IDIM    4096
#define NEXP    8
#define TOPK    2
#define MAXM    (T_TOK * TOPK)   // worst-case token copies per expert
#define BM      32               // token-copy rows per block (2 x 16-row WMMA subtiles)
#define IC      128              // intermediate (I) chunk per pipeline stage
#define XS      1028             // padded LDS row stride for X (floats)
#define AS      132              // padded LDS row stride for act (floats)

typedef float v2f __attribute__((ext_vector_type(2)));
typedef float v8f __attribute__((ext_vector_type(8)));

static __device__ inline v8f wmma4(v2f a, v2f b, v8f c) {
  // D = A(16x4 f32) * B(4x16 f32) + C(16x16 f32)
  return __builtin_amdgcn_wmma_f32_16x16x4_f32(
      /*neg_a=*/false, a, /*neg_b=*/false, b,
      /*c_mod=*/(short)0, c, /*reuse_a=*/false, /*reuse_b=*/false);
}

static __device__ inline float fast_sigmoid(float x) {
  // v_exp_f32 + v_rcp_f32 (avoids the IEEE div_scale refinement chain)
  return __builtin_amdgcn_rcpf(1.0f + __expf(-x));
}

// ---------------- kernel 0: zero output + expert counters ----------------
__global__ void moe_zero_kernel(float* __restrict__ out, int* __restrict__ counts, int n) {
  int i = blockIdx.x * blockDim.x + threadIdx.x;
  if (i < n) out[i] = 0.0f;
  if (i < NEXP) counts[i] = 0;
}

// ---------------- kernel 1: top-2 routing ----------------
// softmax + renormalize over top-2 collapses to sigmoid(g1 - g2).
__global__ void moe_routing_kernel(const float* __restrict__ gates,
                                   int* __restrict__ counts,
                                   int* __restrict__ tok_idx,
                                   float* __restrict__ tok_w) {
  int t = blockIdx.x * blockDim.x + threadIdx.x;
  if (t >= T_TOK) return;
  float g[NEXP];
#pragma unroll
  for (int e = 0; e < NEXP; ++e) g[e] = gates[t * NEXP + e];

  int i1 = 0; float v1 = g[0];
#pragma unroll
  for (int e = 1; e < NEXP; ++e) if (g[e] > v1) { v1 = g[e]; i1 = e; }
  int i2 = (i1 == 0) ? 1 : 0; float v2 = g[i2];
#pragma unroll
  for (int e = 0; e < NEXP; ++e) if (e != i1 && g[e] > v2) { v2 = g[e]; i2 = e; }

  float p1 = fast_sigmoid(v1 - v2);   // renormalized top-1 weight
  float p2 = 1.0f - p1;

  int s1 = atomicAdd(&counts[i1], 1);
  tok_idx[i1 * MAXM + s1] = t;
  tok_w  [i1 * MAXM + s1] = p1;
  int s2 = atomicAdd(&counts[i2], 1);
  tok_idx[i2 * MAXM + s2] = t;
  tok_w  [i2 * MAXM + s2] = p2;
}

// ---------------- kernel 2: fused grouped GEMM1 + SwiGLU + GEMM2 ----------------
// grid.x = expert, grid.y = token tile; 256 threads = 8 waves (wave32).
// Each wave: GEMM1 -> 2 M-subtiles x {gate, up} (B reused 2x);
//            GEMM2 -> 2 M-subtiles x 8 H-tiles  (B reused 2x).
__global__ __launch_bounds__(256)
void moe_expert_kernel(const float* __restrict__ hs,
                       const float* __restrict__ w1,
                       const float* __restrict__ w2,
                       const int*   __restrict__ counts,
                       const int*   __restrict__ tok_idx,
                       const float* __restrict__ tok_w,
                       float* __restrict__ out) {
  extern __shared__ float smem[];
  float* Xs   = smem;                    // [BM][XS]  gathered activations (f32)
  float* As   = Xs + BM * XS;            // [BM][AS]  SwiGLU chunk
  int*   tokS = (int*)(As + BM * AS);    // [BM]
  float* wS   = (float*)(tokS + BM);     // [BM]

  const int e      = blockIdx.x;
  const int mbase  = blockIdx.y * BM;
  const int mcount = counts[e];
  if (mbase >= mcount) return;           // block-uniform exit (EXEC stays full)
  const int rows = (mcount - mbase < BM) ? (mcount - mbase) : BM;

  const int tid  = threadIdx.x;
  const int lane = tid & 31;
  const int wave = tid >> 5;             // 0..7
  const int nidx = lane & 15;            // N (and M for A-operand) index
  const int half = lane >> 4;            // K-pair selector (0 -> K{0,1}, 1 -> K{2,3})

  if (tid < BM) {
    int r = mbase + ((tid < rows) ? tid : 0);
    tokS[tid] = tok_idx[e * MAXM + r];
    wS[tid]   = (tid < rows) ? tok_w[e * MAXM + r] : 0.0f;
  }
  __syncthreads();

  // Gather X tile [BM x HDIM] into LDS (float4 vectorized, conflict-free pad).
  for (int idx = tid; idx < BM * (HDIM / 4); idx += 256) {
    int row = idx >> 8;                  // HDIM/4 == 256
    int c4  = idx & 255;
    const float4 v = ((const float4*)(hs + (size_t)tokS[row] * HDIM))[c4];
    *(float4*)&Xs[row * XS + c4 * 4] = v;
  }
  __syncthreads();

  const float* __restrict__ w1e = w1 + (size_t)e * (2 * IDIM) * HDIM;
  const float* __restrict__ w2e = w2 + (size_t)e * HDIM * IDIM;

  const float* xrow0 = &Xs[nidx * XS + 2 * half];           // A rows 0..15
  const float* xrow1 = &Xs[(nidx + 16) * XS + 2 * half];    // A rows 16..31
  const float* arow0 = &As[nidx * AS + 2 * half];
  const float* arow1 = &As[(nidx + 16) * AS + 2 * half];

  // Per-lane GEMM2 B base pointers: this wave owns H columns [wave*128, wave*128+128)
  const float* w2p[8];
#pragma unroll
  for (int t = 0; t < 8; ++t)
    w2p[t] = w2e + (size_t)(wave * 128 + t * 16 + nidx) * IDIM + 2 * half;

  v8f acc[2][8];
#pragma unroll
  for (int ms = 0; ms < 2; ++ms)
#pragma unroll
    for (int t = 0; t < 8; ++t) acc[ms][t] = (v8f){0,0,0,0,0,0,0,0};

  for (int ic = 0; ic < IDIM; ic += IC) {
    // ---- GEMM1: this wave computes columns [cg, cg+16) of gate and up ----
    const int cg = ic + wave * 16;
    const float* bg = w1e + (size_t)(cg + nidx) * HDIM + 2 * half;          // gate rows
    const float* bu = w1e + (size_t)(IDIM + cg + nidx) * HDIM + 2 * half;   // up rows

    // prefetch next i-chunk's weight rows while we compute on this one
    __builtin_prefetch(bg + (size_t)IC * HDIM, 0, 0);
    __builtin_prefetch(bu + (size_t)IC * HDIM, 0, 0);

    v8f hg0 = (v8f){0,0,0,0,0,0,0,0}, hg1 = hg0, hu0 = hg0, hu1 = hg0;
#pragma unroll 4
    for (int k = 0; k < HDIM; k += 4) {
      v2f a0; a0.x = xrow0[k]; a0.y = xrow0[k + 1];   // ds_load_b64 (rows 0-15)
      v2f a1; a1.x = xrow1[k]; a1.y = xrow1[k + 1];   // ds_load_b64 (rows 16-31)
      v2f b0; b0.x = bg[k];    b0.y = bg[k + 1];      // global b64 (gate), reused 2x
      v2f b1; b1.x = bu[k];    b1.y = bu[k + 1];      // global b64 (up),   reused 2x
      hg0 = wmma4(a0, b0, hg0);
      hg1 = wmma4(a1, b0, hg1);
      hu0 = wmma4(a0, b1, hu0);
      hu1 = wmma4(a1, b1, hu1);
    }

    // ---- SwiGLU: act = silu(hg) * hu, staged to LDS for GEMM2 ----
#pragma unroll
    for (int j = 0; j < 8; ++j) {
      float g0 = hg0[j], g1 = hg1[j];
      int   M  = j + 8 * half;
      As[M * AS + wave * 16 + nidx]        = g0 * fast_sigmoid(g0) * hu0[j];
      As[(M + 16) * AS + wave * 16 + nidx] = g1 * fast_sigmoid(g1) * hu1[j];
    }
    __syncthreads();

    // ---- GEMM2: acc += act_chunk @ w2[:, ic:ic+IC].T ----
#pragma unroll 1
    for (int kk = 0; kk < IC; kk += 4) {
      v2f a0; a0.x = arow0[kk]; a0.y = arow0[kk + 1];
      v2f a1; a1.x = arow1[kk]; a1.y = arow1[kk + 1];
#pragma unroll
      for (int t = 0; t < 8; ++t) {
        v2f b; b.x = w2p[t][ic + kk]; b.y = w2p[t][ic + kk + 1];  // reused 2x
        acc[0][t] = wmma4(a0, b, acc[0][t]);
        acc[1][t] = wmma4(a1, b, acc[1][t]);
      }
    }
    __syncthreads();   // As reused next chunk
  }

  // ---- weighted scatter into the shared output (two experts per token) ----
#pragma unroll
  for (int ms = 0; ms < 2; ++ms) {
#pragma unroll
    for (int t = 0; t < 8; ++t) {
#pragma unroll
      for (int j = 0; j < 8; ++j) {
        int M = j + 8 * half + 16 * ms;
        if (M < rows) {
          int   token = tokS[M];
          float wgt   = wS[M];
          int   hc    = wave * 128 + t * 16 + nidx;
          atomicAdd(&out[(size_t)token * HDIM + hc], wgt * acc[ms][t][j]);
        }
      }
    }
  }
}

// ---------------- launch ----------------
extern "C" void kernel_launch(void* const* d_in, const int* in_sizes, int n_in,
                              void* d_out, int out_size, void* d_ws, size_t ws_size,
                              hipStream_t stream) {
  const float* hs    = (const float*)d_in[0];   // [1024,1024]
  const float* w1    = (const float*)d_in[1];   // [8, 8192, 1024]
  const float* w2    = (const float*)d_in[2];   // [8, 1024, 4096]
  const float* gates = (const float*)d_in[3];   // [1024, 8]
  // d_in[4] = topk (== 2, baked into routing kernel)
  float* out = (float*)d_out;

  // workspace carve-up
  int*   counts  = (int*)d_ws;
  int*   tok_idx = (int*)((char*)d_ws + 256);
  float* tok_w   = (float*)((char*)d_ws + 256 + (size_t)NEXP * MAXM * sizeof(int));

  const int nOut = T_TOK * HDIM;
  moe_zero_kernel<<<(nOut + 255) / 256, 256, 0, stream>>>(out, counts, nOut);
  moe_routing_kernel<<<(T_TOK + 255) / 256, 256, 0, stream>>>(gates, counts, tok_idx, tok_w);

  const size_t smemBytes = (size_t)(BM * XS + BM * AS + BM /*tok*/ + BM /*w*/) * 4;
  dim3 grid(NEXP, MAXM / BM);
  moe_expert_kernel<<<grid, 256, smemBytes, stream>>>(hs, w1, w2, counts, tok_idx, tok_w, out);
}